// SelfFuseGT3D_11115375362872
// MI455X (gfx1250) — compile-verified
//
#include <hip/hip_runtime.h>
#include <hip/hip_bf16.h>

// 26-neighbor context aggregation (3D stencil-weighted gather), fp32.
// B=2, C=32, D=H=W=64, S=1. padded feature is (B,C,66,66,66).
// Memory-bound: ~195 MB compulsory traffic @ 23.3 TB/s => ~8.4 us floor.
// Strategy: async-stage contiguous 3-plane feature slabs into LDS
// (CDNA5 GLOBAL_LOAD_ASYNC_TO_LDS path, ASYNCcnt), reuse each feature
// element ~9x from LDS, and amortize each affinity load across a
// channel pair held in registers.

#define WP     66
#define PLANE  (WP * WP)        // 4356 floats per padded z-plane
#define SLAB   (3 * PLANE)      // 13068 floats = 52272 B (3 planes, 1 channel)
#define CG     2                // channels per block (affinity reuse factor)
#define NSTR   (64 * 64 * 64)   // 262144: stride between n / c planes

#if defined(__has_builtin)
#  if __has_builtin(__builtin_amdgcn_global_load_async_to_lds_b128)
#    define HAVE_ASYNC_LDS 1
#  endif
#  if __has_builtin(__builtin_amdgcn_s_wait_asynccnt)
#    define HAVE_WAIT_ASYNC 1
#  endif
#endif

// Builtin parameter type per hipcc diagnostic:
//   '__attribute__((__vector_size__(4 * sizeof(int)))) int __device__ *'
// i.e. addrspace(1) pointer to a 4 x i32 vector. LDS side: addrspace(3).
typedef int v4i __attribute__((vector_size(16)));
typedef __attribute__((address_space(1))) v4i gv4i_t;
typedef __attribute__((address_space(3))) v4i lv4i_t;

__global__ __launch_bounds__(256) void ctx26_kernel(
    const float* __restrict__ feat,   // (2,32,66,66,66)
    const float* __restrict__ aff,    // (2,26,64,64,64)
    float* __restrict__ out)          // (2,32,64,64,64)
{
    __shared__ float slab[CG * SLAB]; // 104544 bytes

    const int z  = blockIdx.x;        // 0..63
    const int cg = blockIdx.y;        // 0..(32/CG - 1)
    const int b  = blockIdx.z;        // 0..1
    const int tid = threadIdx.x;      // 0..255

    // ---- Stage CG feature slabs. Padded planes z..z+2 are contiguous. ----
#pragma unroll
    for (int cc = 0; cc < CG; ++cc) {
        const int c = cg * CG + cc;
        const float* src = feat + (size_t)((b * 32 + c) * WP + z) * PLANE;
        const v4i* gs = (const v4i*)src;             // 16B-aligned (plane = 17424 B)
        v4i* ds = (v4i*)&slab[cc * SLAB];
#if HAVE_ASYNC_LDS
        for (int i = tid; i < SLAB / 4; i += 256) {
            __builtin_amdgcn_global_load_async_to_lds_b128(
                (gv4i_t*)(gs + i), (lv4i_t*)(ds + i), 0, 0);
        }
#else
        for (int i = tid; i < SLAB / 4; i += 256) {
            ds[i] = gs[i];
        }
#endif
    }
#if HAVE_ASYNC_LDS
#  if HAVE_WAIT_ASYNC
    __builtin_amdgcn_s_wait_asynccnt(0);
#  else
    asm volatile("s_wait_asynccnt 0" ::: "memory");
#  endif
#endif
    __syncthreads();

    // ---- Compute: thread owns column x, 16 strided rows. ----
    const int x  = tid & 63;
    const int y0 = tid >> 6;          // 0..3

    const float* affBase = aff + (size_t)b * 26 * NSTR + (size_t)z * 4096;
    float* outBase = out + (size_t)(b * 32 + cg * CG) * NSTR + (size_t)z * 4096;

    for (int r = 0; r < 16; ++r) {
        const int y = y0 + (r << 2);              // covers 0..63 uniquely
        const float* ap = affBase + (size_t)y * 64 + x;

        float acc0 = 0.0f;
        float acc1 = 0.0f;

        int n = 0;
#pragma unroll
        for (int dz = 0; dz < 3; ++dz) {
#pragma unroll
            for (int dy = 0; dy < 3; ++dy) {
#pragma unroll
                for (int dx = 0; dx < 3; ++dx) {
                    if (dz == 1 && dy == 1 && dx == 1) continue; // skip center
                    const float a = ap[(size_t)n * NSTR];
                    const int off = dz * PLANE + (y + dy) * WP + (x + dx);
                    acc0 = fmaf(slab[off], a, acc0);
                    acc1 = fmaf(slab[SLAB + off], a, acc1);
                    ++n;
                }
            }
        }

        outBase[(size_t)y * 64 + x]        = acc0;
        outBase[NSTR + (size_t)y * 64 + x] = acc1;
    }
}

extern "C" void kernel_launch(void* const* d_in, const int* in_sizes, int n_in,
                              void* d_out, int out_size, void* d_ws, size_t ws_size,
                              hipStream_t stream) {
    (void)in_sizes; (void)n_in; (void)d_ws; (void)ws_size; (void)out_size;
    const float* feat = (const float*)d_in[0];  // padded_feature (2,32,66,66,66) fp32
    const float* aff  = (const float*)d_in[1];  // affinity       (2,26,64,64,64) fp32
    float* out        = (float*)d_out;          // context        (2,32,64,64,64) fp32

    dim3 grid(64, 32 / CG, 2);   // (z, channel-pair, batch)
    dim3 block(256);
    ctx26_kernel<<<grid, block, 0, stream>>>(feat, aff, out);
}